// _HausdorffERLossBase_55327768707885
// MI455X (gfx1250) — compile-verified
//
#include <hip/hip_runtime.h>

// -------------------------------------------------------------------------
// HausdorffERLoss on MI455X (gfx1250).
// Bandwidth-bound: ~670MB of traffic over 10 erosion iterations; both
// ping-pong buffers (64 MiB) fit in the 192MB L2. The 3x3 cross conv is
// expressed per 16x16 tile as 0.2*(T*X + X*T - X) with T tridiagonal-of-ones,
// computed with 8 chained V_WMMA_F32_16X16X4_F32 ops (f32 matched precision),
// plus 4 halo row/col VALU adds. Per-channel min/max via uint atomics
// (order-independent => deterministic); sums via fixed-order tree reductions.
// -------------------------------------------------------------------------

#define HH 512
#define WW 512
#define CC 4
#define HW (HH * WW)
#define NCH 32               // B*C = 8*4 channels
#define BLOCKS_PER_CH 128    // 1024 tiles / 8 waves per block
#define NBLOCKS_A (NCH * BLOCKS_PER_CH)
#define NTOT 8388608.0f      // 8*4*512*512

typedef float v2f __attribute__((ext_vector_type(2)));
typedef float v8f __attribute__((ext_vector_type(8)));

__device__ __forceinline__ float tridiag01(int m, int k) {
    // 1.0 if |m-k| <= 1 else 0.0
    return ((unsigned)(m - k + 1) <= 2u) ? 1.0f : 0.0f;
}

// One wave computes one 16x16 output tile of one (b,c) channel image.
// 8 waves per block; 128 blocks per channel; 4096 blocks total.
__global__ __launch_bounds__(256) void hdloss_conv_kernel(
    const float* __restrict__ pred,     // (8,4,512,512)
    const int*   __restrict__ target,   // (8,1,512,512)
    const float* __restrict__ Eprev,    // raw erosion from previous iter
    float*       __restrict__ Ecur,     // raw erosion out
    const float* __restrict__ normcoef, // [NCH][2] = {scale, off}
    unsigned int* __restrict__ statsMin,
    unsigned int* __restrict__ statsMax,
    float*       __restrict__ partials, // [NBLOCKS_A] per-block raw sums
    int mode)                            // 0: bound=(pred-onehot)^2, 1: normalize Eprev
{
    __shared__ float lds[8][18 * 19];   // per-wave 18x18 haloed tile, stride 19
    __shared__ float wsum[8], wmax[8], wmin[8];

    const int wave = threadIdx.x >> 5;
    const int lane = threadIdx.x & 31;
    const int ch = blockIdx.x >> 7;            // 128 blocks per channel
    const int blockInCh = blockIdx.x & 127;
    const int tile = blockInCh * 8 + wave;     // 0..1023
    const int r0 = (tile >> 5) * 16;
    const int c0 = (tile & 31) * 16;
    const int b = ch >> 2;
    const int c = ch & 3;

    float scale = 1.0f, off = 0.0f;
    if (mode) { scale = normcoef[2 * ch]; off = normcoef[2 * ch + 1]; }

    // ---- Stage 18x18 haloed tile into LDS (zero-fill outside image) ----
    float* tileL = lds[wave];
    for (int i = lane; i < 18 * 18; i += 32) {
        const int rr = i / 18, cc2 = i % 18;
        const int gr = r0 + rr - 1, gc = c0 + cc2 - 1;
        float v = 0.0f;
        if (gr >= 0 && gr < HH && gc >= 0 && gc < WW) {
            const int idx = ch * HW + gr * WW + gc;
            if (mode) {
                v = (Eprev[idx] - off) * scale;
            } else {
                const float p = pred[idx];
                const int t = target[(b * HH + gr) * WW + gc];
                const float d = p - ((t == c) ? 1.0f : 0.0f);
                v = d * d;
            }
        }
        tileL[rr * 19 + cc2] = v;
    }
    __syncthreads();

    // ---- acc = T*X + X*T via 8x V_WMMA_F32_16X16X4_F32 ----
    const int lh = lane >> 4;     // lane half selects k-pair / M+8
    const int l16 = lane & 15;
    v8f acc = {0.f, 0.f, 0.f, 0.f, 0.f, 0.f, 0.f, 0.f};

#pragma unroll
    for (int s = 0; s < 4; ++s) {           // T*X : A = T, B = X rows kb..
        const int kb = 4 * s + 2 * lh;
        v2f ta, xb;
        ta[0] = tridiag01(l16, kb);
        ta[1] = tridiag01(l16, kb + 1);
        xb[0] = tileL[(kb + 1) * 19 + (l16 + 1)];
        xb[1] = tileL[(kb + 2) * 19 + (l16 + 1)];
        acc = __builtin_amdgcn_wmma_f32_16x16x4_f32(
            false, ta, false, xb, (short)0, acc, false, false);
    }
#pragma unroll
    for (int s = 0; s < 4; ++s) {           // X*T : A = X cols kb.., B = T
        const int kb = 4 * s + 2 * lh;
        v2f xa, tb;
        xa[0] = tileL[(l16 + 1) * 19 + (kb + 1)];
        xa[1] = tileL[(l16 + 1) * 19 + (kb + 2)];
        tb[0] = tridiag01(kb, l16);
        tb[1] = tridiag01(kb + 1, l16);
        acc = __builtin_amdgcn_wmma_f32_16x16x4_f32(
            false, xa, false, tb, (short)0, acc, false, false);
    }

    // ---- erosion = relu(0.2*(acc - x + halos) - 0.5); store + reduce ----
    float lsum = 0.0f, lmax = 0.0f, lmin = __builtin_inff();
#pragma unroll
    for (int r = 0; r < 8; ++r) {
        const int m = r + 8 * lh;
        const int n = l16;
        float h = acc[r] - tileL[(m + 1) * 19 + (n + 1)];
        if (m == 0)  h += tileL[0 * 19 + (n + 1)];
        if (m == 15) h += tileL[17 * 19 + (n + 1)];
        if (n == 0)  h += tileL[(m + 1) * 19 + 0];
        if (n == 15) h += tileL[(m + 1) * 19 + 17];
        const float e = fmaxf(0.2f * h - 0.5f, 0.0f);
        Ecur[ch * HW + (r0 + m) * WW + (c0 + n)] = e;
        lsum += e;
        lmax = fmaxf(lmax, e);
        lmin = fminf(lmin, e);
    }

    // wave32 butterfly reduction
#pragma unroll
    for (int o = 16; o > 0; o >>= 1) {
        lsum += __shfl_xor(lsum, o, 32);
        lmax = fmaxf(lmax, __shfl_xor(lmax, o, 32));
        lmin = fminf(lmin, __shfl_xor(lmin, o, 32));
    }
    if (lane == 0) { wsum[wave] = lsum; wmax[wave] = lmax; wmin[wave] = lmin; }
    __syncthreads();
    if (threadIdx.x == 0) {
        float bs = 0.0f, bM = 0.0f, bm = __builtin_inff();
        for (int w = 0; w < 8; ++w) {
            bs += wsum[w];
            bM = fmaxf(bM, wmax[w]);
            bm = fminf(bm, wmin[w]);
        }
        partials[blockIdx.x] = bs;                         // fixed-order later
        atomicMax(&statsMax[ch], __float_as_uint(bM));     // e >= 0: uint order ok
        atomicMin(&statsMin[ch], __float_as_uint(bm));
    }
}

__global__ void hdloss_init_kernel(unsigned int* statsMin, unsigned int* statsMax,
                                   float* normcoef, float* lossacc) {
    const int i = threadIdx.x;
    if (i < NCH) {
        statsMin[i] = 0x7F800000u;  // +inf
        statsMax[i] = 0u;
        normcoef[2 * i] = 1.0f;
        normcoef[2 * i + 1] = 0.0f;
        lossacc[i] = 0.0f;
    }
}

// One block per channel: deterministic tree-reduce of 128 partial sums,
// fold normalization analytically, accumulate weighted loss, reset stats.
__global__ __launch_bounds__(128) void hdloss_finish_kernel(
    const float* __restrict__ partials,
    unsigned int* __restrict__ statsMin, unsigned int* __restrict__ statsMax,
    float* __restrict__ normcoef, float* __restrict__ lossacc, float wk)
{
    __shared__ float red[128];
    const int ch = blockIdx.x;
    red[threadIdx.x] = partials[ch * BLOCKS_PER_CH + threadIdx.x];
    __syncthreads();
    for (int o = 64; o > 0; o >>= 1) {
        if (threadIdx.x < o) red[threadIdx.x] += red[threadIdx.x + o];
        __syncthreads();
    }
    if (threadIdx.x == 0) {
        const float sum = red[0];
        const float mn = __uint_as_float(statsMin[ch]);
        const float mx = __uint_as_float(statsMax[ch]);
        const float den = mx - mn;
        float snorm, scl, offv;
        if (den != 0.0f) {
            snorm = (sum - (float)HW * mn) / den;  // sum of (e-mn)/den
            scl = 1.0f / den;
            offv = mn;
        } else {
            snorm = sum; scl = 1.0f; offv = 0.0f;
        }
        lossacc[ch] += wk * snorm;
        normcoef[2 * ch] = scl;
        normcoef[2 * ch + 1] = offv;
        statsMin[ch] = 0x7F800000u;   // reset for next iteration
        statsMax[ch] = 0u;
    }
}

__global__ void hdloss_final_kernel(const float* __restrict__ lossacc,
                                    float* __restrict__ out) {
    if (threadIdx.x == 0) {
        float s = 0.0f;
        for (int i = 0; i < NCH; ++i) s += lossacc[i];
        out[0] = s / NTOT;
    }
}

extern "C" void kernel_launch(void* const* d_in, const int* in_sizes, int n_in,
                              void* d_out, int out_size, void* d_ws, size_t ws_size,
                              hipStream_t stream) {
    const float* pred   = (const float*)d_in[0];
    const int*   target = (const int*)d_in[1];

    char* ws = (char*)d_ws;
    size_t o = 0;
    float* Ea = (float*)(ws + o); o += (size_t)HW * NCH * sizeof(float);   // 32 MiB
    float* Eb = (float*)(ws + o); o += (size_t)HW * NCH * sizeof(float);   // 32 MiB
    float* partials = (float*)(ws + o); o += NBLOCKS_A * sizeof(float);
    unsigned int* statsMin = (unsigned int*)(ws + o); o += NCH * sizeof(unsigned int);
    unsigned int* statsMax = (unsigned int*)(ws + o); o += NCH * sizeof(unsigned int);
    float* normcoef = (float*)(ws + o); o += 2 * NCH * sizeof(float);
    float* lossacc = (float*)(ws + o); o += NCH * sizeof(float);
    (void)ws_size; (void)in_sizes; (void)n_in; (void)out_size;

    hdloss_init_kernel<<<1, 32, 0, stream>>>(statsMin, statsMax, normcoef, lossacc);

    for (int k = 0; k < 10; ++k) {
        float* Ecur = (k & 1) ? Eb : Ea;
        const float* Eprev = (k & 1) ? Ea : Eb;
        const float wk = (float)((k + 1) * (k + 1));   // (k+1)^ALPHA, ALPHA=2
        hdloss_conv_kernel<<<NBLOCKS_A, 256, 0, stream>>>(
            pred, target, Eprev, Ecur, normcoef, statsMin, statsMax, partials,
            (k == 0) ? 0 : 1);
        hdloss_finish_kernel<<<NCH, 128, 0, stream>>>(
            partials, statsMin, statsMax, normcoef, lossacc, wk);
    }

    hdloss_final_kernel<<<1, 32, 0, stream>>>(lossacc, (float*)d_out);
}